// FeaturePreprocess_2662879723940
// MI455X (gfx1250) — compile-verified
//
#include <hip/hip_runtime.h>
#include <hip/hip_bf16.h>
#include <math.h>

// ---------------------------------------------------------------------------
// FeaturePreprocess for MI455X (gfx1250, wave32, WMMA)
// Pass 1: single-sweep global moments (S1, S2, cnt) over 26.7M f32 using
//         V_WMMA_F32_16X16X4_F32 as the accumulation engine (B = ones).
// Pass 2: deterministic f64 combine -> mean/std, then tiny gather/resample.
// ---------------------------------------------------------------------------

typedef __attribute__((ext_vector_type(2))) float v2f;
typedef __attribute__((ext_vector_type(8))) float v8f;

#define WG_SIZE      256
#define N_BLOCKS     1024
#define N_LM         543
#define ROWF         1629          // 543 * 3
#define N_SEL        107
#define FIXED_FRAMES 30

__constant__ int c_sel[N_SEL] = {
    // LIPS (40)
    61, 185, 40, 39, 37, 0, 267, 269, 270, 409, 291,
    146, 91, 181, 84, 17, 314, 405, 321, 375,
    78, 191, 80, 81, 82, 13, 312, 311, 310, 415,
    95, 88, 178, 87, 14, 317, 402, 318, 324, 308,
    // LEFT_HAND (21): 468..488
    468, 469, 470, 471, 472, 473, 474, 475, 476, 477, 478,
    479, 480, 481, 482, 483, 484, 485, 486, 487, 488,
    // UPPER_BODY (25): 489..513
    489, 490, 491, 492, 493, 494, 495, 496, 497, 498, 499, 500, 501,
    502, 503, 504, 505, 506, 507, 508, 509, 510, 511, 512, 513,
    // RIGHT_HAND (21): 522..542
    522, 523, 524, 525, 526, 527, 528, 529, 530, 531, 532,
    533, 534, 535, 536, 537, 538, 539, 540, 541, 542
};

__device__ __forceinline__ v8f wmma_acc(v2f a, v2f b, v8f c) {
    // D = A(16x4 f32) x B(4x16 f32) + C ; B = ones => D[M][j] = C[M][j] + sum_K A[M][K]
    return __builtin_amdgcn_wmma_f32_16x16x4_f32(
        /*neg_a=*/false, a, /*neg_b=*/false, b,
        /*c_mod=*/(short)0, c, /*reuse_a=*/false, /*reuse_b=*/false);
}

__global__ __launch_bounds__(WG_SIZE)
void fp_reduce_kernel(const float* __restrict__ x,
                      float* __restrict__ partials,
                      long long N) {
    const int tid       = threadIdx.x;
    const int lane      = tid & 31;
    const int waveInBlk = tid >> 5;

    const long long gwave   = (long long)blockIdx.x * (WG_SIZE / 32) + waveInBlk;
    const long long nWaves  = (long long)gridDim.x * (WG_SIZE / 32);
    const long long N4      = N >> 2;        // float4 count
    const long long nChunks = N4 >> 5;       // 32 float4 per wave per chunk (128 floats)

    const float4* __restrict__ x4 = (const float4*)x;

    v8f c_sum = {};
    v8f c_sq  = {};
    v8f c_cn  = {};
    v2f ones; ones[0] = 1.0f; ones[1] = 1.0f;

    // Loop condition is uniform per wave -> EXEC is all-ones at every WMMA.
    for (long long ch = gwave; ch < nChunks; ch += nWaves) {
        const long long i = (ch << 5) + lane;
        float4 v = x4[i];
        __builtin_prefetch((const void*)(x4 + ((ch + nWaves) << 5) + lane), 0, 1);

        const float k0 = (v.x == v.x) ? 1.0f : 0.0f;
        const float k1 = (v.y == v.y) ? 1.0f : 0.0f;
        const float k2 = (v.z == v.z) ? 1.0f : 0.0f;
        const float k3 = (v.w == v.w) ? 1.0f : 0.0f;
        const float x0 = k0 ? v.x : 0.0f;
        const float x1 = k1 ? v.y : 0.0f;
        const float x2 = k2 ? v.z : 0.0f;
        const float x3 = k3 ? v.w : 0.0f;

        v2f aS0; aS0[0] = x0;      aS0[1] = x1;
        v2f aS1; aS1[0] = x2;      aS1[1] = x3;
        v2f aQ0; aQ0[0] = x0 * x0; aQ0[1] = x1 * x1;
        v2f aQ1; aQ1[0] = x2 * x2; aQ1[1] = x3 * x3;
        v2f aC0; aC0[0] = k0;      aC0[1] = k1;
        v2f aC1; aC1[0] = k2;      aC1[1] = k3;

        c_sum = wmma_acc(aS0, ones, c_sum);
        c_sq  = wmma_acc(aQ0, ones, c_sq);
        c_cn  = wmma_acc(aC0, ones, c_cn);
        c_sum = wmma_acc(aS1, ones, c_sum);
        c_sq  = wmma_acc(aQ1, ones, c_sq);
        c_cn  = wmma_acc(aC1, ones, c_cn);
    }

    // Per-lane: sum my 8 accumulator rows (rows 0-7 of col j for lane j<16,
    // rows 8-15 of col j for lane j+16). Column sum = full reduction result.
    float tS = ((c_sum[0] + c_sum[1]) + (c_sum[2] + c_sum[3])) +
               ((c_sum[4] + c_sum[5]) + (c_sum[6] + c_sum[7]));
    float tQ = ((c_sq[0]  + c_sq[1])  + (c_sq[2]  + c_sq[3]))  +
               ((c_sq[4]  + c_sq[5])  + (c_sq[6]  + c_sq[7]));
    float tC = ((c_cn[0]  + c_cn[1])  + (c_cn[2]  + c_cn[3]))  +
               ((c_cn[4]  + c_cn[5])  + (c_cn[6]  + c_cn[7]));
    tS += __shfl_xor(tS, 16, 32);
    tQ += __shfl_xor(tQ, 16, 32);
    tC += __shfl_xor(tC, 16, 32);

    __shared__ float red[3][WG_SIZE / 32];
    if (lane == 0) {
        red[0][waveInBlk] = tS;
        red[1][waveInBlk] = tQ;
        red[2][waveInBlk] = tC;
    }
    __syncthreads();

    if (tid == 0) {
        float S = 0.0f, Q = 0.0f, C = 0.0f;
        #pragma unroll
        for (int w = 0; w < WG_SIZE / 32; ++w) {
            S += red[0][w]; Q += red[1][w]; C += red[2][w];
        }
        if (blockIdx.x == 0) {
            // scalar tail: elements not covered by full 128-float wave chunks
            for (long long i = nChunks << 7; i < N; ++i) {
                float v = x[i];
                if (v == v) { S += v; Q += v * v; C += 1.0f; }
            }
        }
        partials[(size_t)blockIdx.x * 3 + 0] = S;
        partials[(size_t)blockIdx.x * 3 + 1] = Q;
        partials[(size_t)blockIdx.x * 3 + 2] = C;
    }
}

__global__ __launch_bounds__(256)
void fp_finalize_kernel(const float* __restrict__ x,
                        const float* __restrict__ partials,
                        int nParts,
                        float* __restrict__ out,
                        int T) {
    __shared__ double sd[3][256];
    const int tid = threadIdx.x;

    double S = 0.0, Q = 0.0, C = 0.0;
    for (int i = tid; i < nParts; i += 256) {
        S += (double)partials[i * 3 + 0];
        Q += (double)partials[i * 3 + 1];
        C += (double)partials[i * 3 + 2];
    }
    sd[0][tid] = S; sd[1][tid] = Q; sd[2][tid] = C;
    __syncthreads();
    for (int off = 128; off > 0; off >>= 1) {
        if (tid < off) {
            sd[0][tid] += sd[0][tid + off];
            sd[1][tid] += sd[1][tid + off];
            sd[2][tid] += sd[2][tid + off];
        }
        __syncthreads();
    }

    __shared__ float s_mean, s_std;
    if (tid == 0) {
        const double S1  = sd[0][0];
        const double S2  = sd[1][0];
        const double cnt = sd[2][0];
        const double mean = S1 / cnt;
        // reference: m2 = sum(xc)/cnt ; var = sum((xc - m2)^2)/(cnt-1)
        const double Sc1 = S1 - cnt * mean;                      // sum(x - mean)
        const double m2c = Sc1 / cnt;
        const double Sc2 = S2 - 2.0 * mean * S1 + cnt * mean * mean;  // sum((x-mean)^2)
        const double var = (Sc2 - 2.0 * m2c * Sc1 + cnt * m2c * m2c) / (cnt - 1.0);
        s_mean = (float)mean;
        s_std  = (float)sqrt(var);
    }
    __syncthreads();
    const float mean = s_mean;
    const float stdv = s_std;

    const int total = FIXED_FRAMES * N_SEL * 3;
    const float scale = (float)T / (float)FIXED_FRAMES;
    for (int i = tid; i < total; i += 256) {
        const int f  = i / (N_SEL * 3);
        const int r  = i % (N_SEL * 3);
        const int l  = r / 3;
        const int cc = r % 3;
        const int land = c_sel[l];
        float val;
        if (T >= FIXED_FRAMES) {
            // nearest-exact, f32 math as in the reference
            int fi = (int)floorf(((float)f + 0.5f) * scale);
            fi = fi < 0 ? 0 : (fi > T - 1 ? T - 1 : fi);
            const float v = x[(size_t)fi * ROWF + land * 3 + cc];
            val = (v == v) ? (v - mean) / stdv : 0.0f;
        } else {
            // linear, align_corners=False
            float pos = ((float)f + 0.5f) * scale - 0.5f;
            pos = fminf(fmaxf(pos, 0.0f), (float)(T - 1));
            const int i0 = (int)floorf(pos);
            const int i1 = (i0 + 1 < T - 1) ? (i0 + 1) : (T - 1);
            const float w = pos - (float)i0;
            const float v0 = x[(size_t)i0 * ROWF + land * 3 + cc];
            const float v1 = x[(size_t)i1 * ROWF + land * 3 + cc];
            const float y0 = (v0 == v0) ? (v0 - mean) / stdv : 0.0f;
            const float y1 = (v1 == v1) ? (v1 - mean) / stdv : 0.0f;
            val = (1.0f - w) * y0 + w * y1;
        }
        out[i] = val;
    }
}

extern "C" void kernel_launch(void* const* d_in, const int* in_sizes, int n_in,
                              void* d_out, int out_size, void* d_ws, size_t ws_size,
                              hipStream_t stream) {
    const float* x = (const float*)d_in[0];
    const long long N = (long long)in_sizes[0];      // T * 543 * 3
    const int T = (int)(N / ROWF);

    float* partials = (float*)d_ws;                  // N_BLOCKS * 3 floats (12 KB)

    fp_reduce_kernel<<<N_BLOCKS, WG_SIZE, 0, stream>>>(x, partials, N);
    fp_finalize_kernel<<<1, 256, 0, stream>>>(x, partials, N_BLOCKS,
                                              (float*)d_out, T);
}